// AdjacencyLearning_38397007626720
// MI455X (gfx1250) — compile-verified
//
#include <hip/hip_runtime.h>
#include <hip/hip_bf16.h>

typedef __attribute__((ext_vector_type(2))) float v2f;
typedef __attribute__((ext_vector_type(8))) float v8f;

#define B_    128
#define R_    36
#define W_    60
#define D_    1024
#define TK_   20
#define EPS_  1e-12f
#define NEG_  -1e30f

// ---------------------------------------------------------------------------
// Kernel 1: per-row inverse L2 norms for cap_words (7680 rows) and
// img_regions (4608 rows). One 256-thread block per 1024-float row.
// ---------------------------------------------------------------------------
__global__ __launch_bounds__(256)
void row_invnorm_kernel(const float* __restrict__ img,
                        const float* __restrict__ cap,
                        float* __restrict__ inv_cap,
                        float* __restrict__ inv_img) {
    __shared__ float red[256];
    const int row = blockIdx.x;
    const float* p = (row < B_ * W_) ? (cap + (size_t)row * D_)
                                     : (img + (size_t)(row - B_ * W_) * D_);
    const float4 v = ((const float4*)p)[threadIdx.x];
    red[threadIdx.x] = v.x * v.x + v.y * v.y + v.z * v.z + v.w * v.w;
    __syncthreads();
    for (int st = 128; st >= 1; st >>= 1) {
        if (threadIdx.x < st) red[threadIdx.x] += red[threadIdx.x + st];
        __syncthreads();
    }
    if (threadIdx.x == 0) {
        const float inv = 1.0f / fmaxf(sqrtf(red[0]), EPS_);
        if (row < B_ * W_) inv_cap[row] = inv;
        else               inv_img[row - B_ * W_] = inv;
    }
}

// ---------------------------------------------------------------------------
// Kernel 2: one block per (caption c, image i) pair.
// 128 threads = 4 waves. Each wave computes a 16x48 band of the 60x36 sim
// tile (padded to 64x48) with v_wmma_f32_16x16x4_f32, K staged through LDS
// in chunks of 32. Then all reductions + top-k + MLPs in LDS.
// ---------------------------------------------------------------------------
__global__ __launch_bounds__(128)
void pair_kernel(const float* __restrict__ img,
                 const float* __restrict__ cap,
                 const float* __restrict__ inv_cap,
                 const float* __restrict__ inv_img,
                 const float* __restrict__ w1_t2i, const float* __restrict__ b1_t2i,
                 const float* __restrict__ w2_t2i, const float* __restrict__ b2_t2i,
                 const float* __restrict__ w1_i2t, const float* __restrict__ b1_i2t,
                 const float* __restrict__ w2_i2t, const float* __restrict__ b2_i2t,
                 const int*   __restrict__ cap_len,
                 float*       __restrict__ out) {
    const int i    = blockIdx.x;   // image index
    const int c    = blockIdx.y;   // caption index
    const int tid  = threadIdx.x;
    const int lane = tid & 31;
    const int wv   = tid >> 5;     // wave id 0..3

    __shared__ float As[64][33];   // cap chunk  (rows 60..63 zero)
    __shared__ float Bs[48][33];   // img chunk  (rows 36..47 zero)
    __shared__ float Sm[64][49];   // sim tile after leaky-relu
    __shared__ float rnw[48];      // 1/norm over words, per region r
    __shared__ float rnr[64];      // 1/norm over regions, per word w
    __shared__ float rowsim[64];
    __shared__ float colsim[48];
    __shared__ float topr[TK_], topc[TK_];
    __shared__ float h1[TK_], h2[TK_];
    __shared__ float red[128];
    __shared__ float s_rowmean, s_colmean;

    // zero the pad rows once (K-loop only rewrites real rows)
    for (int idx = tid; idx < 4 * 33;  idx += 128) As[60 + idx / 33][idx % 33] = 0.f;
    for (int idx = tid; idx < 12 * 33; idx += 128) Bs[36 + idx / 33][idx % 33] = 0.f;

    v8f acc0 = {}; v8f acc1 = {}; v8f acc2 = {};
    const int mr = (wv << 4) + (lane & 15);   // A row for this lane
    const int nr = lane & 15;                 // B/C column within 16-tile
    const int kb = (lane >> 4) << 1;          // K sub-offset {0,2}

    const float* capRow = cap + (size_t)c * W_ * D_;
    const float* imgRow = img + (size_t)i * R_ * D_;
    const float* invC   = inv_cap + c * W_;
    const float* invI   = inv_img + i * R_;

    for (int k0 = 0; k0 < D_; k0 += 32) {
        __syncthreads();
        // stage 60x32 cap + 36x32 img floats, normalized, into LDS
        for (int idx = tid; idx < 480 + 288; idx += 128) {
            if (idx < 480) {
                const int row = idx >> 3, c4 = (idx & 7) << 2;
                const float4 v = *(const float4*)(capRow + (size_t)row * D_ + k0 + c4);
                const float sc = invC[row];
                As[row][c4 + 0] = v.x * sc; As[row][c4 + 1] = v.y * sc;
                As[row][c4 + 2] = v.z * sc; As[row][c4 + 3] = v.w * sc;
            } else {
                const int j = idx - 480;
                const int row = j >> 3, c4 = (j & 7) << 2;
                const float4 v = *(const float4*)(imgRow + (size_t)row * D_ + k0 + c4);
                const float sc = invI[row];
                Bs[row][c4 + 0] = v.x * sc; Bs[row][c4 + 1] = v.y * sc;
                Bs[row][c4 + 2] = v.z * sc; Bs[row][c4 + 3] = v.w * sc;
            }
        }
        __syncthreads();
        // 8 K-steps of 4, 3 N-tiles each -> 24 WMMAs per chunk per wave
        for (int kk = 0; kk < 32; kk += 4) {
            v2f a;   a.x   = As[mr][kk + kb];      a.y   = As[mr][kk + kb + 1];
            v2f b0;  b0.x  = Bs[nr][kk + kb];      b0.y  = Bs[nr][kk + kb + 1];
            v2f b1v; b1v.x = Bs[16 + nr][kk + kb]; b1v.y = Bs[16 + nr][kk + kb + 1];
            v2f b2v; b2v.x = Bs[32 + nr][kk + kb]; b2v.y = Bs[32 + nr][kk + kb + 1];
            acc0 = __builtin_amdgcn_wmma_f32_16x16x4_f32(false, a, false, b0,  (short)0, acc0, false, false);
            acc1 = __builtin_amdgcn_wmma_f32_16x16x4_f32(false, a, false, b1v, (short)0, acc1, false, false);
            acc2 = __builtin_amdgcn_wmma_f32_16x16x4_f32(false, a, false, b2v, (short)0, acc2, false, false);
        }
    }
    __syncthreads();

    // spill accumulators to LDS with leaky-relu.
    // C/D layout: VGPR v, lanes 0-15 -> M=v, lanes 16-31 -> M=v+8; N=lane%16
    {
        const int mb = (wv << 4) + ((lane >> 4) << 3);
        for (int v = 0; v < 8; v++) {
            const int m = mb + v;
            float x0 = acc0[v]; Sm[m][nr]      = (x0 >= 0.f) ? x0 : 0.1f * x0;
            float x1 = acc1[v]; Sm[m][16 + nr] = (x1 >= 0.f) ? x1 : 0.1f * x1;
            float x2 = acc2[v]; Sm[m][32 + nr] = (x2 >= 0.f) ? x2 : 0.1f * x2;
        }
    }
    __syncthreads();

    const int cl   = cap_len[c];
    const int nsel = (cl > TK_) ? cl : TK_;

    // reciprocal norms: over words (per region) and over regions (per word)
    if (tid < R_) {
        float ss = 0.f;
        for (int w = 0; w < nsel; w++) { float x = Sm[w][tid]; ss += x * x; }
        rnw[tid] = 1.0f / fmaxf(sqrtf(ss), EPS_);
    } else if (tid >= 64 && tid < 64 + W_) {
        const int w = tid - 64;
        float ss = 0.f;
        for (int r = 0; r < R_; r++) { float x = Sm[w][r]; ss += x * x; }
        rnr[w] = 1.0f / fmaxf(sqrtf(ss), EPS_);
    }
    __syncthreads();

    // row max (t2i) and col max (i2t)
    if (tid < W_) {
        float m = NEG_;
        if (tid < nsel)
            for (int r = 0; r < R_; r++) m = fmaxf(m, Sm[tid][r] * rnw[r]);
        rowsim[tid] = m;
    } else if (tid >= 64 && tid < 64 + R_) {
        const int r = tid - 64;
        float m = NEG_;
        for (int w = 0; w < cl; w++) m = fmaxf(m, Sm[w][r] * rnr[w]);
        colsim[r] = m;
    }
    __syncthreads();

    // means
    red[tid] = (tid < cl) ? rowsim[tid] : 0.f;
    __syncthreads();
    for (int st = 64; st >= 1; st >>= 1) {
        if (tid < st) red[tid] += red[tid + st];
        __syncthreads();
    }
    if (tid == 0) s_rowmean = red[0] / (float)cl;
    __syncthreads();
    red[tid] = (tid < R_) ? colsim[tid] : 0.f;
    __syncthreads();
    for (int st = 64; st >= 1; st >>= 1) {
        if (tid < st) red[tid] += red[tid + st];
        __syncthreads();
    }
    if (tid == 0) s_colmean = red[0] / (float)R_;
    __syncthreads();

    // top-20 by parallel rank computation (stable, ties broken by index)
    if (tid < W_) {
        const float v = rowsim[tid];
        int rank = 0;
        for (int j = 0; j < W_; j++) {
            const float u = rowsim[j];
            rank += (u > v) || (u == v && j < tid);
        }
        if (rank < TK_) topr[rank] = v;
    } else if (tid >= 64 && tid < 64 + R_) {
        const int r = tid - 64;
        const float v = colsim[r];
        int rank = 0;
        for (int j = 0; j < R_; j++) {
            const float u = colsim[j];
            rank += (u > v) || (u == v && j < r);
        }
        if (rank < TK_) topc[rank] = v;
    }
    __syncthreads();

    // 20->20 hidden layers (t2i on wave0, i2t on wave1)
    if (tid < TK_) {
        float a = b1_t2i[tid];
        for (int k = 0; k < TK_; k++) a += topr[k] * w1_t2i[k * TK_ + tid];
        h1[tid] = fmaxf(a, 0.f);
    } else if (tid >= 32 && tid < 32 + TK_) {
        const int j = tid - 32;
        float a = b1_i2t[j];
        for (int k = 0; k < TK_; k++) a += topc[k] * w1_i2t[k * TK_ + j];
        h2[j] = fmaxf(a, 0.f);
    }
    __syncthreads();

    if (tid == 0) {
        float o = b2_t2i[0];
        for (int j = 0; j < TK_; j++) o += h1[j] * w2_t2i[j];
        out[c * B_ + i]              = o + s_rowmean;   // t2i_rel (C,I)
        out[2 * B_ * B_ + c * B_ + i] = s_rowmean;      // batch_t2i_mean (C,I)
    } else if (tid == 32) {
        float o = b2_i2t[0];
        for (int j = 0; j < TK_; j++) o += h2[j] * w2_i2t[j];
        out[B_ * B_ + i * B_ + c]     = o + s_colmean;  // i2t_rel (I,C)
        out[3 * B_ * B_ + i * B_ + c] = s_colmean;      // batch_i2t_mean (I,C)
    }
}

// ---------------------------------------------------------------------------
extern "C" void kernel_launch(void* const* d_in, const int* in_sizes, int n_in,
                              void* d_out, int out_size, void* d_ws, size_t ws_size,
                              hipStream_t stream) {
    const float* img     = (const float*)d_in[0];
    const float* cap     = (const float*)d_in[1];
    const float* w1_t2i  = (const float*)d_in[2];
    const float* b1_t2i  = (const float*)d_in[3];
    const float* w2_t2i  = (const float*)d_in[4];
    const float* b2_t2i  = (const float*)d_in[5];
    const float* w1_i2t  = (const float*)d_in[6];
    const float* b1_i2t  = (const float*)d_in[7];
    const float* w2_i2t  = (const float*)d_in[8];
    const float* b2_i2t  = (const float*)d_in[9];
    const int*   cap_len = (const int*)d_in[10];
    float*       out     = (float*)d_out;

    float* inv_cap = (float*)d_ws;          // 7680 floats
    float* inv_img = inv_cap + B_ * W_;     // 4608 floats

    row_invnorm_kernel<<<B_ * W_ + B_ * R_, 256, 0, stream>>>(img, cap, inv_cap, inv_img);

    dim3 grid(B_, B_);
    pair_kernel<<<grid, 128, 0, stream>>>(img, cap, inv_cap, inv_img,
                                          w1_t2i, b1_t2i, w2_t2i, b2_t2i,
                                          w1_i2t, b1_i2t, w2_i2t, b2_i2t,
                                          cap_len, out);
}